// graphAttention_61211873902595
// MI455X (gfx1250) — compile-verified
//
#include <hip/hip_runtime.h>
#include <math.h>

typedef __attribute__((ext_vector_type(2))) float v2f;
typedef __attribute__((ext_vector_type(8))) float v8f;

#define DD 256
#define BNEPS 1e-6f

__device__ __forceinline__ float lrelu(float x){ return x > 0.f ? x : 0.01f * x; }
__device__ __forceinline__ float elu1(float x){ return x > 0.f ? x : (expf(x) - 1.f); }
__device__ __forceinline__ float sigm(float x){ return 1.f / (1.f + expf(-x)); }

// order-preserving float<->uint encoding for atomicMax-based segment max
__device__ __forceinline__ unsigned fenc(float f){
    unsigned u = __float_as_uint(f);
    return (u & 0x80000000u) ? ~u : (u | 0x80000000u);
}
__device__ __forceinline__ float fdec(unsigned u){
    return __uint_as_float((u & 0x80000000u) ? (u ^ 0x80000000u) : ~u);
}

// ---------------- K0: zero accumulators (must run every call) ----------------
__global__ void k_init(float* __restrict__ ctx, float* __restrict__ denom,
                       unsigned* __restrict__ segmax, int n_atom){
    int stride = gridDim.x * blockDim.x;
    int total  = n_atom * DD;
    for (int t = blockIdx.x * blockDim.x + threadIdx.x; t < total; t += stride){
        ctx[t] = 0.f;
        if (t < n_atom){ denom[t] = 0.f; segmax[t] = 0u; }
    }
}

// ---------------- K1: per-atom align projections p1 = a.w1, p2 = a.w2 --------
__global__ void __launch_bounds__(256) k_align_proj(
    const float* __restrict__ atom, const float* __restrict__ Wal,
    float* __restrict__ p1, float* __restrict__ p2, int n_atom){
    int wid  = (blockIdx.x * blockDim.x + threadIdx.x) >> 5;
    int lane = threadIdx.x & 31;
    if (wid >= n_atom) return;
    const float* a = atom + (size_t)wid * DD;
    float s1 = 0.f, s2 = 0.f;
    #pragma unroll
    for (int k = lane; k < DD; k += 32){
        float av = a[k];
        s1 += av * Wal[k];
        s2 += av * Wal[DD + k];
    }
    #pragma unroll
    for (int off = 16; off; off >>= 1){
        s1 += __shfl_xor(s1, off, 32);
        s2 += __shfl_xor(s2, off, 32);
    }
    if (lane == 0){ p1[wid] = s1; p2[wid] = s2; }
}

// ---------------- K2: H = BN(atom @ W_att^T + b_att)  (fp32 WMMA) ------------
__global__ void __launch_bounds__(256) k_attend_gemm(
    const float* __restrict__ atom, const float* __restrict__ W,
    const float* __restrict__ bias,
    const float* __restrict__ g, const float* __restrict__ bb,
    const float* __restrict__ mm, const float* __restrict__ vv,
    float* __restrict__ H, int n_atom){
    int wid  = (blockIdx.x * blockDim.x + threadIdx.x) >> 5;
    int lane = threadIdx.x & 31;
    int mtiles = (n_atom + 15) >> 4;
    int mtile = wid >> 4;         // D/16 == 16 n-tiles
    int ntile = wid & 15;
    if (mtile >= mtiles) return;
    int m0 = mtile << 4, n0 = ntile << 4;
    int r  = lane & 15, hv = lane >> 4;

    int arow = m0 + r; if (arow >= n_atom) arow = n_atom - 1;
    const float* ap = atom + (size_t)arow * DD + hv * 2;  // A[m][k..k+1]
    int j = n0 + r;
    const float* bp = W + (size_t)j * DD + hv * 2;        // B[k][j] = W[j][k]

    v8f acc = {};
    #pragma unroll 8
    for (int k = 0; k < DD; k += 4){
        v2f a = *(const v2f*)(ap + k);
        v2f b = *(const v2f*)(bp + k);
        acc = __builtin_amdgcn_wmma_f32_16x16x4_f32(false, a, false, b,
                                                    (short)0, acc, false, false);
    }
    float sc = g[j] * rsqrtf(vv[j] + BNEPS);
    float sh = bb[j] - mm[j] * sc;
    float bj = bias[j];
    #pragma unroll
    for (int t = 0; t < 8; ++t){
        int row = m0 + t + (hv << 3);
        if (row < n_atom) H[(size_t)row * DD + j] = (acc[t] + bj) * sc + sh;
    }
}

// ---------------- K3: edge score + segment max -------------------------------
__global__ void k_edge_score(const int* __restrict__ bi,
    const float* __restrict__ p1, const float* __restrict__ p2,
    const float* __restrict__ b_al, const float* __restrict__ g,
    const float* __restrict__ b, const float* __restrict__ m,
    const float* __restrict__ v,
    float* __restrict__ sbuf, unsigned* __restrict__ segmax, int n_edge){
    int e = blockIdx.x * blockDim.x + threadIdx.x;
    if (e >= n_edge) return;
    int src = bi[e], dst = bi[n_edge + e];
    float s  = p1[src] + p2[dst] + b_al[0];
    float sc = g[0] * rsqrtf(v[0] + BNEPS);
    s = (s - m[0]) * sc + b[0];
    s = lrelu(s);
    sbuf[e] = s;
    atomicMax(segmax + src, fenc(s));
}

// ---------------- K4: exp(s - max) + segment sum -----------------------------
__global__ void k_edge_exp(const int* __restrict__ bi,
    const unsigned* __restrict__ segmax, float* __restrict__ sbuf,
    float* __restrict__ denom, int n_edge){
    int e = blockIdx.x * blockDim.x + threadIdx.x;
    if (e >= n_edge) return;
    int src = bi[e];
    float val = expf(sbuf[e] - fdec(segmax[src]));
    sbuf[e] = val;
    atomicAdd(denom + src, val);
}

// ---------------- K5: context[src] += attn * H[dst] (one wave per edge) ------
__global__ void __launch_bounds__(256) k_scatter(const int* __restrict__ bi,
    const float* __restrict__ sbuf, const float* __restrict__ denom,
    const float* __restrict__ H, float* __restrict__ ctx, int n_edge){
    int wid  = (blockIdx.x * blockDim.x + threadIdx.x) >> 5;
    int lane = threadIdx.x & 31;
    if (wid >= n_edge) return;
    int src = bi[wid], dst = bi[n_edge + wid];
    float attn = sbuf[wid] / (denom[src] + 1e-8f);
    const float* h = H + (size_t)dst * DD;
    float* c = ctx + (size_t)src * DD;
    #pragma unroll
    for (int j = 0; j < DD; j += 32)
        atomicAdd(c + j + lane, attn * h[j + lane]);
}

// ---------------- K6: fused ELU + GRUCell (6 WMMA accumulators) --------------
__global__ void __launch_bounds__(256) k_gru(
    const float* __restrict__ atom, const float* __restrict__ ctx,
    const float* __restrict__ Wih, const float* __restrict__ Whh,
    const float* __restrict__ bih, const float* __restrict__ bhh,
    float* __restrict__ out, int n_atom){
    int wid  = (blockIdx.x * blockDim.x + threadIdx.x) >> 5;
    int lane = threadIdx.x & 31;
    int mtiles = (n_atom + 15) >> 4;
    int mtile = wid >> 4;
    int ntile = wid & 15;
    if (mtile >= mtiles) return;
    int m0 = mtile << 4, n0 = ntile << 4;
    int r  = lane & 15, hv = lane >> 4;

    int arow = m0 + r; if (arow >= n_atom) arow = n_atom - 1;
    const float* ac = ctx  + (size_t)arow * DD + hv * 2;
    const float* ah = atom + (size_t)arow * DD + hv * 2;
    int j = n0 + r;
    const float* b_ir = Wih + (size_t)(j         ) * DD + hv * 2;
    const float* b_iz = Wih + (size_t)(j +     DD) * DD + hv * 2;
    const float* b_in = Wih + (size_t)(j + 2 * DD) * DD + hv * 2;
    const float* b_hr = Whh + (size_t)(j         ) * DD + hv * 2;
    const float* b_hz = Whh + (size_t)(j +     DD) * DD + hv * 2;
    const float* b_hn = Whh + (size_t)(j + 2 * DD) * DD + hv * 2;

    v8f cir = {}, ciz = {}, cin = {}, dhr = {}, dhz = {}, dhn = {};
    #pragma unroll 4
    for (int k = 0; k < DD; k += 4){
        v2f a1 = *(const v2f*)(ac + k);
        a1.x = elu1(a1.x); a1.y = elu1(a1.y);     // ELU(context) on the fly
        v2f a2 = *(const v2f*)(ah + k);
        v2f w;
        w = *(const v2f*)(b_ir + k);
        cir = __builtin_amdgcn_wmma_f32_16x16x4_f32(false, a1, false, w, (short)0, cir, false, false);
        w = *(const v2f*)(b_iz + k);
        ciz = __builtin_amdgcn_wmma_f32_16x16x4_f32(false, a1, false, w, (short)0, ciz, false, false);
        w = *(const v2f*)(b_in + k);
        cin = __builtin_amdgcn_wmma_f32_16x16x4_f32(false, a1, false, w, (short)0, cin, false, false);
        w = *(const v2f*)(b_hr + k);
        dhr = __builtin_amdgcn_wmma_f32_16x16x4_f32(false, a2, false, w, (short)0, dhr, false, false);
        w = *(const v2f*)(b_hz + k);
        dhz = __builtin_amdgcn_wmma_f32_16x16x4_f32(false, a2, false, w, (short)0, dhz, false, false);
        w = *(const v2f*)(b_hn + k);
        dhn = __builtin_amdgcn_wmma_f32_16x16x4_f32(false, a2, false, w, (short)0, dhn, false, false);
    }
    float Bir = bih[j], Biz = bih[j + DD], Bin = bih[j + 2 * DD];
    float Bhr = bhh[j], Bhz = bhh[j + DD], Bhn = bhh[j + 2 * DD];
    #pragma unroll
    for (int t = 0; t < 8; ++t){
        int row = m0 + t + (hv << 3);
        float rg = sigm((cir[t] + Bir) + (dhr[t] + Bhr));
        float zg = sigm((ciz[t] + Biz) + (dhz[t] + Bhz));
        float ng = tanhf((cin[t] + Bin) + rg * (dhn[t] + Bhn));
        if (row < n_atom){
            float a0 = atom[(size_t)row * DD + j];
            out[(size_t)row * DD + j] = (1.f - zg) * ng + zg * a0;
        }
    }
}

extern "C" void kernel_launch(void* const* d_in, const int* in_sizes, int n_in,
                              void* d_out, int out_size, void* d_ws, size_t ws_size,
                              hipStream_t stream){
    const float* atom = (const float*)d_in[0];
    const int*   bi   = (const int*)  d_in[1];
    const float* Wal  = (const float*)d_in[2];
    const float* b_al = (const float*)d_in[3];
    const float* bag  = (const float*)d_in[4];
    const float* bab  = (const float*)d_in[5];
    const float* bam  = (const float*)d_in[6];
    const float* bav  = (const float*)d_in[7];
    const float* Watt = (const float*)d_in[8];
    const float* batt = (const float*)d_in[9];
    const float* btg  = (const float*)d_in[10];
    const float* btb  = (const float*)d_in[11];
    const float* btm  = (const float*)d_in[12];
    const float* btv  = (const float*)d_in[13];
    const float* Wih  = (const float*)d_in[14];
    const float* Whh  = (const float*)d_in[15];
    const float* bih  = (const float*)d_in[16];
    const float* bhh  = (const float*)d_in[17];
    float* out = (float*)d_out;

    int n_atom = in_sizes[0] / DD;
    int n_edge = in_sizes[1] / 2;

    // workspace layout (floats): p1 | p2 | denom | segmax | sbuf | H | ctx
    float*    ws     = (float*)d_ws;
    float*    p1     = ws;
    float*    p2     = p1 + n_atom;
    float*    denom  = p2 + n_atom;
    unsigned* segmax = (unsigned*)(denom + n_atom);
    float*    sbuf   = (float*)(segmax + n_atom);
    float*    H      = sbuf + n_edge;
    float*    ctx    = H + (size_t)n_atom * DD;

    int mtiles = (n_atom + 15) / 16;
    int waves  = mtiles * (DD / 16);
    int nbT    = (waves + 7) / 8;           // 8 waves / 256-thread block

    k_init<<<1024, 256, 0, stream>>>(ctx, denom, segmax, n_atom);
    k_align_proj<<<(n_atom + 7) / 8, 256, 0, stream>>>(atom, Wal, p1, p2, n_atom);
    k_attend_gemm<<<nbT, 256, 0, stream>>>(atom, Watt, batt, btg, btb, btm, btv, H, n_atom);
    k_edge_score<<<(n_edge + 255) / 256, 256, 0, stream>>>(bi, p1, p2, b_al, bag, bab, bam, bav,
                                                           sbuf, segmax, n_edge);
    k_edge_exp<<<(n_edge + 255) / 256, 256, 0, stream>>>(bi, segmax, sbuf, denom, n_edge);
    k_scatter<<<(n_edge + 7) / 8, 256, 0, stream>>>(bi, sbuf, denom, H, ctx, n_edge);
    k_gru<<<nbT, 256, 0, stream>>>(atom, ctx, Wih, Whh, bih, bhh, out, n_atom);
}